// MultiHeadSparseAttention_24558622998726
// MI455X (gfx1250) — compile-verified
//
#include <hip/hip_runtime.h>
#include <hip/hip_bf16.h>

#define B_    2
#define S_    2048
#define D_    512
#define H_    8
#define DH_   64
#define BH_   16
#define TOPK_ 64
#define NEGV  (-1000000.0f)

typedef __attribute__((ext_vector_type(2))) float    v2f;
typedef __attribute__((ext_vector_type(8))) float    v8f;
typedef __attribute__((ext_vector_type(4))) unsigned v4u;
typedef __attribute__((ext_vector_type(4))) int      v4i;
typedef __attribute__((ext_vector_type(8))) int      v8i;

#if defined(__has_builtin)
#if __has_builtin(__builtin_amdgcn_tensor_load_to_lds)
#define USE_TDM 1
#endif
#endif
#ifndef USE_TDM
#define USE_TDM 0
#endif

// Order-preserving float -> uint key (total order incl. negatives).
__device__ __forceinline__ unsigned fkey(float f) {
  unsigned u = __float_as_uint(f);
  return (u & 0x80000000u) ? ~u : (u | 0x80000000u);
}

// Raw LDS byte offset of a __shared__ object (flat LDS addr keeps offset in
// the low 32 bits; upper bits are the aperture).
__device__ __forceinline__ unsigned lds_addr_of(const void* p) {
  return (unsigned)(uintptr_t)p;
}

#if USE_TDM
// Issue a 2-D TDM tile load: tile_d1 rows of tile_d0 elements (fp32),
// row stride = stride_elems, LDS destination padded by pad_amount DWORDs
// every 2^(pad_interval_code+1) DWORDs (matches our padded LDS arrays).
// Must be executed from a SCALAR branch by exactly one wave (EXEC ignored).
__device__ __forceinline__ void tdm_load_2d(unsigned lds_byte, const void* gptr,
                                            unsigned tile_d0, unsigned tile_d1,
                                            unsigned long long stride_elems,
                                            unsigned tensor_d0, unsigned tensor_d1,
                                            unsigned pad_interval_code,
                                            unsigned pad_amount_code) {
  unsigned long long ga = (unsigned long long)(uintptr_t)gptr;
  v4u g0;
  g0.x = 1u;                                             // count=1 (user desc)
  g0.y = lds_byte;                                       // lds_addr
  g0.z = (unsigned)(ga & 0xFFFFFFFFull);                 // global_addr[31:0]
  g0.w = (unsigned)((ga >> 32) & 0x01FFFFFFull)          // global_addr[56:32]
         | 0x80000000u;                                  // type=2 ("image")
  v8i g1;
  g1[0] = (int)((2u << 16)                               // data_size = 4B
                | (1u << 20)                             // pad_enable
                | (pad_interval_code << 22) | (pad_amount_code << 25));
  g1[1] = (int)((tensor_d0 & 0xFFFFu) << 16);            // tensor_dim0 lo16
  g1[2] = (int)((tensor_d0 >> 16) | ((tensor_d1 & 0xFFFFu) << 16));
  g1[3] = (int)((tensor_d1 >> 16) | (tile_d0 << 16));    // tile_dim0
  g1[4] = (int)tile_d1;                                  // tile_dim1 (dim2=0)
  g1[5] = (int)(stride_elems & 0xFFFFFFFFull);           // dim0 stride lo
  g1[6] = (int)(stride_elems >> 32);                     // dim0 stride hi
  g1[7] = 0;
  v4i z4 = {0, 0, 0, 0};
#if __clang_major__ >= 23
  v8i z8 = {0, 0, 0, 0, 0, 0, 0, 0};
  __builtin_amdgcn_tensor_load_to_lds(g0, g1, z4, z4, z8, 0);
#else
  __builtin_amdgcn_tensor_load_to_lds(g0, g1, z4, z4, 0);
#endif
}
#endif  // USE_TDM

// ---------------------------------------------------------------------------
// GEMM: Y = X @ W, X is [4096 x 512], W is [512 x 512].
// MODE 0: X row-major, Y scattered into head layout [B*H][S][DH].
// MODE 1: X gathered from head layout,  Y row-major [4096][512].
// fp32 WMMA 16x16x4, 8 waves, WG tile 128(M) x 64(N), wave tile 32x32.
// Operand tiles double-buffered in LDS; TDM for chunk n+1 overlaps WMMA on n.
// ---------------------------------------------------------------------------
template <int MODE>
__global__ __launch_bounds__(256)
void gemm512_wmma(const float* __restrict__ X, const float* __restrict__ W,
                  float* __restrict__ Y) {
  __shared__ float sX[2][128][17];  // 16-DW rows + 1 DW pad (TDM codes 3,0)
  __shared__ float sW[2][16][68];   // 64-DW rows + 4 DW pad (TDM codes 5,3)

  const int tid  = threadIdx.x;
  const int lane = tid & 31;
  const int wv   = tid >> 5;   // wave 0..7
  const int wm   = wv >> 1;    // 0..3 : M direction (32 rows each)
  const int wn   = wv & 1;     // 0..1 : N direction (32 cols each)
  const int lr   = lane & 15;
  const int lh   = lane >> 4;  // wave half (selects K 0,1 vs 2,3 in frags)

  const int rowBase = blockIdx.y * 128;
  const int colBase = blockIdx.x * 64;

#if USE_TDM
  auto issueTiles = [&](int kb, int buf) {
    const float* xsrc;
    unsigned long long xstride;
    if (MODE == 0) {
      xsrc = X + (size_t)rowBase * D_ + kb;
      xstride = D_;
    } else {
      int bb = rowBase >> 11, ss = rowBase & (S_ - 1);
      int hh = kb >> 6;
      xsrc = X + ((size_t)(bb * H_ + hh) * S_ + ss) * DH_ + (kb & 63);
      xstride = DH_;
    }
    tdm_load_2d(lds_addr_of(&sX[buf][0][0]), xsrc, 16u, 128u, xstride,
                (MODE == 0) ? 512u : 64u, 4096u, 3u, 0u);
    tdm_load_2d(lds_addr_of(&sW[buf][0][0]), W + (size_t)kb * D_ + colBase,
                64u, 16u, D_, 512u, 512u, 5u, 3u);
  };
#endif

  v8f acc[2][2] = {};

#if USE_TDM
  if (__builtin_amdgcn_readfirstlane(wv) == 0) issueTiles(0, 0);
#endif

  for (int t = 0; t < 32; ++t) {
    const int cur = t & 1;
    const int kb  = t * 16;
    __syncthreads();   // previous compute done -> next buffer reusable
#if USE_TDM
    if (__builtin_amdgcn_readfirstlane(wv) == 0) {   // scalar branch: one wave
      if (t + 1 < 32) {
        issueTiles((t + 1) * 16, cur ^ 1);
        __builtin_amdgcn_s_wait_tensorcnt(2);   // tiles for t done, t+1 in flight
      } else {
        __builtin_amdgcn_s_wait_tensorcnt(0);
      }
    }
#else
    {
      int r  = tid >> 1;
      int c0 = (tid & 1) * 8;
      const float* src;
      if (MODE == 0) {
        src = X + (rowBase + r) * D_ + kb + c0;
      } else {
        int m  = rowBase + r;
        int bb = m >> 11, ss = m & (S_ - 1);
        int hh = kb >> 6;
        src = X + ((bb * H_ + hh) * S_ + ss) * DH_ + (kb & 63) + c0;
      }
#pragma unroll
      for (int i = 0; i < 8; ++i) sX[cur][r][c0 + i] = src[i];
    }
    {
      int r  = tid >> 4;
      int c0 = (tid & 15) * 4;
      const float* src = W + (kb + r) * D_ + colBase + c0;
#pragma unroll
      for (int i = 0; i < 4; ++i) sW[cur][r][c0 + i] = src[i];
    }
#endif
    __syncthreads();   // buffer `cur` published

#pragma unroll
    for (int kk = 0; kk < 16; kk += 4) {
      v2f af[2], bf[2];
#pragma unroll
      for (int mt = 0; mt < 2; ++mt) {
        int m = wm * 32 + mt * 16 + lr;
        af[mt].x = sX[cur][m][kk + 2 * lh + 0];
        af[mt].y = sX[cur][m][kk + 2 * lh + 1];
      }
#pragma unroll
      for (int nt = 0; nt < 2; ++nt) {
        int n = wn * 32 + nt * 16 + lr;
        bf[nt].x = sW[cur][kk + 2 * lh + 0][n];
        bf[nt].y = sW[cur][kk + 2 * lh + 1][n];
      }
#pragma unroll
      for (int mt = 0; mt < 2; ++mt)
#pragma unroll
        for (int nt = 0; nt < 2; ++nt)
          acc[mt][nt] = __builtin_amdgcn_wmma_f32_16x16x4_f32(
              false, af[mt], false, bf[nt], (short)0, acc[mt][nt], false, false);
    }
  }

  // ---- epilogue: C/D layout VGPR i -> M = i (+8 for upper lane half) ----
#pragma unroll
  for (int mt = 0; mt < 2; ++mt) {
#pragma unroll
    for (int nt = 0; nt < 2; ++nt) {
#pragma unroll
      for (int i = 0; i < 8; ++i) {
        int m = rowBase + wm * 32 + mt * 16 + i + lh * 8;
        int c = colBase + wn * 32 + nt * 16 + lr;
        float v = acc[mt][nt][i];
        if (MODE == 0) {
          int bb = m >> 11, ss = m & (S_ - 1);
          int hh = c >> 6, dd = c & 63;
          Y[((bb * H_ + hh) * S_ + ss) * DH_ + dd] = v;
        } else {
          Y[m * D_ + c] = v;
        }
      }
    }
  }
}

// ---------------------------------------------------------------------------
// Sparse attention: one WG per (head-batch, 16-query block).
// Phase 1: scores = (Q K^T)/8 with valid-len mask -> LDS [16][2048]
// Phase 2: exact top-64 threshold (bisection on uint keys) + softmax
// Phase 3: out = P @ V  (K-split across wave pairs, LDS reduce)
// K/V streamed in 128-key chunks via double-buffered TDM; V chunk 0's DMA is
// issued before phase 2 so it overlaps the top-k work.
// ---------------------------------------------------------------------------
__global__ __launch_bounds__(256)
void attn_topk_wmma(const float* __restrict__ Q, const float* __restrict__ K,
                    const float* __restrict__ V,
                    const long long* __restrict__ valid_lens,
                    float* __restrict__ O) {
  __shared__ float sScore[16][2049];  // stride 2049: conflict-free columns
  __shared__ float sKV[2][128][65];   // 128 keys x 64 dims (TDM codes 5,0)
  __shared__ float sQ[16][65];        // (TDM codes 5,0)
  __shared__ float sRed[4][256];

  const int tid  = threadIdx.x;
  const int lane = tid & 31;
  const int wv   = tid >> 5;
  const int lr   = lane & 15;
  const int lh   = lane >> 4;

  const int qb = blockIdx.x * 16;
  const int bh = blockIdx.y;
  const int vl = (int)valid_lens[bh / H_];

  const float* Qp = Q + (size_t)(bh * S_ + qb) * DH_;
  const float* Kp = K + (size_t)bh * S_ * DH_;
  const float* Vp = V + (size_t)bh * S_ * DH_;

#if USE_TDM
  if (__builtin_amdgcn_readfirstlane(wv) == 0) {
    // Q tile (16 contiguous rows of 64) + first K chunk.
    tdm_load_2d(lds_addr_of(&sQ[0][0]), Qp, 64u, 16u, DH_, 64u, 2048u, 5u, 0u);
    tdm_load_2d(lds_addr_of(&sKV[0][0][0]), Kp, 64u, 128u, DH_, 64u, 2048u, 5u, 0u);
  }
#else
  {
    int r = tid >> 4, c = (tid & 15) * 4;
#pragma unroll
    for (int i = 0; i < 4; ++i) sQ[r][c + i] = Qp[r * DH_ + c + i];
  }
#endif

  // ---------------- phase 1: masked, scaled scores ----------------
  for (int ch = 0; ch < 16; ++ch) {
    const int cur = ch & 1;
    __syncthreads();   // compute on previous chunk done
#if USE_TDM
    if (__builtin_amdgcn_readfirstlane(wv) == 0) {
      if (ch + 1 < 16) {
        tdm_load_2d(lds_addr_of(&sKV[cur ^ 1][0][0]),
                    Kp + (size_t)(ch + 1) * 128 * DH_,
                    64u, 128u, DH_, 64u, 2048u, 5u, 0u);
        __builtin_amdgcn_s_wait_tensorcnt(1);   // chunk ch ready, ch+1 in flight
      } else {
        __builtin_amdgcn_s_wait_tensorcnt(0);
      }
    }
#else
    {
      int r = tid >> 1, c = (tid & 1) * 32;
      const float* src = Kp + (ch * 128 + r) * DH_ + c;
#pragma unroll
      for (int i = 0; i < 32; ++i) sKV[cur][r][c + i] = src[i];
    }
#endif
    __syncthreads();
    {
      v8f c8a = {}, c8b = {};
      int kl = wv * 16;                      // all 8 waves: one key block each
#pragma unroll
      for (int d0 = 0; d0 < DH_; d0 += 8) {  // two independent C chains
        v2f a, b;
        a.x = sQ[lr][d0 + 2 * lh + 0];
        a.y = sQ[lr][d0 + 2 * lh + 1];
        b.x = sKV[cur][kl + lr][d0 + 2 * lh + 0];   // B[d][n] = K[key=n][d]
        b.y = sKV[cur][kl + lr][d0 + 2 * lh + 1];
        c8a = __builtin_amdgcn_wmma_f32_16x16x4_f32(
            false, a, false, b, (short)0, c8a, false, false);
        a.x = sQ[lr][d0 + 4 + 2 * lh + 0];
        a.y = sQ[lr][d0 + 4 + 2 * lh + 1];
        b.x = sKV[cur][kl + lr][d0 + 4 + 2 * lh + 0];
        b.y = sKV[cur][kl + lr][d0 + 4 + 2 * lh + 1];
        c8b = __builtin_amdgcn_wmma_f32_16x16x4_f32(
            false, a, false, b, (short)0, c8b, false, false);
      }
      v8f c8 = c8a + c8b;
      int col = ch * 128 + kl + lr;
#pragma unroll
      for (int i = 0; i < 8; ++i) {
        float sc = c8[i] * 0.125f;   // 1/sqrt(DH)
        sScore[i + lh * 8][col] = (col < vl) ? sc : NEGV;
      }
    }
  }
  __syncthreads();

#if USE_TDM
  // Prefetch first V chunk; the DMA overlaps the whole top-k/softmax phase.
  if (__builtin_amdgcn_readfirstlane(wv) == 0) {
    tdm_load_2d(lds_addr_of(&sKV[0][0][0]), Vp, 64u, 128u, DH_, 64u, 2048u, 5u, 0u);
  }
#endif

  // ------------- phase 2: exact top-64 + masked softmax -------------
#pragma unroll 1
  for (int rr = 0; rr < 2; ++rr) {
    int row = wv + rr * 8;
    // largest t with count(key >= t) >= TOPK  ==  key of 64th-largest score
    unsigned lo = 0u, hi = 0xFFFFFFFFu;
    while (lo < hi) {
      unsigned mid = (unsigned)(((unsigned long long)lo + hi + 1ull) >> 1);
      int cnt = 0;
      for (int c = lane; c < S_; c += 32)
        cnt += (fkey(sScore[row][c]) >= mid) ? 1 : 0;
#pragma unroll
      for (int off = 16; off > 0; off >>= 1) cnt += __shfl_xor(cnt, off, 32);
      if (cnt >= TOPK_) lo = mid; else hi = mid - 1u;
    }
    const unsigned kth = lo;

    float mx = -3.0e38f;
    for (int c = lane; c < S_; c += 32) mx = fmaxf(mx, sScore[row][c]);
#pragma unroll
    for (int off = 16; off > 0; off >>= 1) mx = fmaxf(mx, __shfl_xor(mx, off, 32));

    float sum = 0.f;
    for (int c = lane; c < S_; c += 32) {
      float s = sScore[row][c];
      float p = (fkey(s) >= kth) ? __expf(s - mx) : 0.f;  // dropped -> 0
      sScore[row][c] = p;
      sum += p;
    }
#pragma unroll
    for (int off = 16; off > 0; off >>= 1) sum += __shfl_xor(sum, off, 32);
    float inv = 1.0f / sum;
    for (int c = lane; c < S_; c += 32) sScore[row][c] *= inv;
  }
  __syncthreads();

  // ---------------- phase 3: out = P @ V ----------------
  const int ntile = wv & 3;   // which 16-wide slice of DH
  const int sub   = wv >> 2;  // K-split half of each 128-key chunk
  v8f acc = {}, acc2 = {};
  for (int ch = 0; ch < 16; ++ch) {
    const int cur = ch & 1;
    __syncthreads();
#if USE_TDM
    if (__builtin_amdgcn_readfirstlane(wv) == 0) {
      if (ch + 1 < 16) {
        tdm_load_2d(lds_addr_of(&sKV[cur ^ 1][0][0]),
                    Vp + (size_t)(ch + 1) * 128 * DH_,
                    64u, 128u, DH_, 64u, 2048u, 5u, 0u);
        __builtin_amdgcn_s_wait_tensorcnt(1);
      } else {
        __builtin_amdgcn_s_wait_tensorcnt(0);
      }
    }
#else
    {
      int r = tid >> 1, c = (tid & 1) * 32;
      const float* src = Vp + (ch * 128 + r) * DH_ + c;
#pragma unroll
      for (int i = 0; i < 32; ++i) sKV[cur][r][c + i] = src[i];
    }
#endif
    __syncthreads();
    int l0 = sub * 64;
    int gbase = ch * 128 + l0;
#pragma unroll
    for (int kk = 0; kk < 64; kk += 8) {     // two independent C chains
      v2f a, b;
      a.x = sScore[lr][gbase + kk + 2 * lh + 0];               // A = P (16 x 4)
      a.y = sScore[lr][gbase + kk + 2 * lh + 1];
      b.x = sKV[cur][l0 + kk + 2 * lh + 0][ntile * 16 + lr];   // B = V (4 x 16)
      b.y = sKV[cur][l0 + kk + 2 * lh + 1][ntile * 16 + lr];
      acc = __builtin_amdgcn_wmma_f32_16x16x4_f32(
          false, a, false, b, (short)0, acc, false, false);
      a.x = sScore[lr][gbase + kk + 4 + 2 * lh + 0];
      a.y = sScore[lr][gbase + kk + 4 + 2 * lh + 1];
      b.x = sKV[cur][l0 + kk + 4 + 2 * lh + 0][ntile * 16 + lr];
      b.y = sKV[cur][l0 + kk + 4 + 2 * lh + 1][ntile * 16 + lr];
      acc2 = __builtin_amdgcn_wmma_f32_16x16x4_f32(
          false, a, false, b, (short)0, acc2, false, false);
    }
  }
  acc = acc + acc2;
  __syncthreads();
  if (sub == 1) {
#pragma unroll
    for (int i = 0; i < 8; ++i) sRed[ntile][i * 32 + lane] = acc[i];
  }
  __syncthreads();
  if (sub == 0) {
#pragma unroll
    for (int i = 0; i < 8; ++i) {
      float v = acc[i] + sRed[ntile][i * 32 + lane];
      O[(size_t)(bh * S_ + qb + i + lh * 8) * DH_ + ntile * 16 + lr] = v;
    }
  }
}

// ---------------------------------------------------------------------------
extern "C" void kernel_launch(void* const* d_in, const int* in_sizes, int n_in,
                              void* d_out, int out_size, void* d_ws, size_t ws_size,
                              hipStream_t stream) {
  (void)in_sizes; (void)n_in; (void)out_size; (void)ws_size;
  const float*     q_in = (const float*)d_in[0];
  const float*     k_in = (const float*)d_in[1];
  const float*     v_in = (const float*)d_in[2];
  const long long* vls  = (const long long*)d_in[3];   // int64 valid_lens
  const float*     Wq   = (const float*)d_in[4];
  const float*     Wk   = (const float*)d_in[5];
  const float*     Wv   = (const float*)d_in[6];
  const float*     Wo   = (const float*)d_in[7];
  float*           out  = (float*)d_out;

  // workspace: head-split Q, K, V, attn-out, each B*H*S*DH = 2,097,152 floats
  const size_t seg = (size_t)BH_ * S_ * DH_;
  float* ws = (float*)d_ws;
  float* Qh = ws;
  float* Kh = ws + seg;
  float* Vh = ws + 2 * seg;
  float* Ah = ws + 3 * seg;

  dim3 blk(256, 1, 1);
  dim3 gGrid(D_ / 64, (B_ * S_) / 128, 1);   // (8, 32)

  gemm512_wmma<0><<<gGrid, blk, 0, stream>>>(q_in, Wq, Qh);
  gemm512_wmma<0><<<gGrid, blk, 0, stream>>>(k_in, Wk, Kh);
  gemm512_wmma<0><<<gGrid, blk, 0, stream>>>(v_in, Wv, Vh);

  attn_topk_wmma<<<dim3(S_ / 16, BH_, 1), blk, 0, stream>>>(Qh, Kh, Vh, vls, Ah);

  gemm512_wmma<1><<<gGrid, blk, 0, stream>>>(Ah, Wo, out);
}